// Classifier_74792560492870
// MI455X (gfx1250) — compile-verified
//
#include <hip/hip_runtime.h>
#include <hip/hip_bf16.h>

typedef __attribute__((ext_vector_type(2))) float v2f;
typedef __attribute__((ext_vector_type(8))) float v8f;

#define NGRAPH 512
#define NODES  200
#define EDGES  6400
#define HID    128
#define NCLS   10
#define PITCH  132      // t-buffer pitch (floats): stride%64==4 -> conflict-free A reads
#define APITCH 204      // adjacency pitch: stride%64==12, gcd(12,64)=4 -> 16 distinct banks
#define ROWSP  208      // 200 padded to 13 M-tiles of 16

// ---------------------------------------------------------------------------
// Fused per-graph GNN, one workgroup (8 waves) per graph:
//   degrees + dense-adjacency build (LDS atomics, 3 per edge)
//   gconv1 collapses to per-node scalars (rank-1) -> GEMM1 A synthesized in VALU
//   t = (h1*inv_out) @ W2            : WMMA f32 16x16x4, B strip in registers
//   agg = Adj @ t                    : WMMA f32 16x16x4 (densified scatter!)
//   h2 = relu(agg*inv_in + b2), mean-pool in registers -> hg[g]
// ---------------------------------------------------------------------------
__global__ __launch_bounds__(256) void gnn_fused_kernel(
    const int* __restrict__ src, const int* __restrict__ dst,
    const float* __restrict__ W1, const float* __restrict__ b1,
    const float* __restrict__ W2, const float* __restrict__ b2,
    float* __restrict__ hg)
{
  __shared__ float adj [ROWSP * APITCH];  // 169,728 B dense adjacency (counts)
  __shared__ float tbuf[ROWSP * PITCH];   // 109,824 B GEMM1 output
  __shared__ float degIn[ROWSP], degOut[ROWSP];
  __shared__ float sval[ROWSP], agg1[ROWSP], invIn[ROWSP], pArr[ROWSP], qArr[ROWSP];
  __shared__ float w1s[HID], b1s[HID];

  const int g    = blockIdx.x;
  const int tid  = threadIdx.x;
  const int lane = tid & 31;
  const int wave = tid >> 5;
  const int lmod = lane & 15;
  const int lhi  = lane >> 4;          // 0 or 1

  __builtin_prefetch(&W2[tid * 64], 0, 1);  // warm L2 (global_prefetch_b8)

  // ---- phase 0: zero adjacency (b128 stores), degrees; stage W1/b1 -------
  float4 z4 = {0.f, 0.f, 0.f, 0.f};
  float4* adj4 = (float4*)adj;
  for (int i = tid; i < ROWSP * APITCH / 4; i += 256) adj4[i] = z4;
  for (int n = tid; n < ROWSP; n += 256) { degIn[n] = 0.0f; degOut[n] = 0.0f; }
  if (tid < HID) { w1s[tid] = W1[tid]; b1s[tid] = b1[tid]; }
  __syncthreads();

  // ---- phase 1: one edge pass: degrees + dense adjacency -----------------
  const int ebase = g * EDGES;
  for (int e = tid; e < EDGES; e += 256) {
    int sl = src[ebase + e] - g * NODES;
    int dl = dst[ebase + e] - g * NODES;
    atomicAdd(&degOut[sl], 1.0f);
    atomicAdd(&degIn[dl], 1.0f);
    atomicAdd(&adj[dl * APITCH + sl], 1.0f);   // Adj[dst][src] += 1
  }
  __syncthreads();

  // ---- phase 2: per-node scale factors -----------------------------------
  for (int n = tid; n < ROWSP; n += 256) {
    if (n < NODES) {
      float io = rsqrtf(fmaxf(degOut[n], 1.0f));
      float ii = rsqrtf(fmaxf(degIn[n], 1.0f));
      qArr[n]  = io;                 // inv_out
      invIn[n] = ii;
      sval[n]  = degIn[n] * io;      // h0 * inv_out (gconv1 is rank-1)
    } else {
      qArr[n] = 0.0f; invIn[n] = 0.0f; sval[n] = 0.0f;  // pad rows -> A=0
    }
    agg1[n] = 0.0f;
  }
  __syncthreads();

  // ---- phase 3: gconv1 scalar edge aggregation ---------------------------
  for (int e = tid; e < EDGES; e += 256) {
    int sl = src[ebase + e] - g * NODES;
    int dl = dst[ebase + e] - g * NODES;
    atomicAdd(&agg1[dl], sval[sl]);
  }
  __syncthreads();
  for (int n = tid; n < ROWSP; n += 256) pArr[n] = agg1[n] * invIn[n];
  __syncthreads();

  // ---- phase 4: GEMM1: t = (relu(p*W1+b1)*q) @ W2 ------------------------
  // wave j owns channels [j*16, j*16+16); W2 strip in 64 regs; A on the fly.
  float breg[32][2];
#pragma unroll
  for (int kk = 0; kk < 32; ++kk) {
#pragma unroll
    for (int k = 0; k < 2; ++k) {
      int krow = kk * 4 + 2 * lhi + k;
      breg[kk][k] = W2[krow * HID + wave * 16 + lmod];
    }
  }
  for (int i = 0; i < ROWSP / 16; ++i) {
    int   n  = i * 16 + lmod;
    float pn = pArr[n];
    float qn = qArr[n];
    v8f acc = {0.f, 0.f, 0.f, 0.f, 0.f, 0.f, 0.f, 0.f};
#pragma unroll
    for (int kk = 0; kk < 32; ++kk) {
      int kcol = kk * 4 + 2 * lhi;
      v2f a;
      a.x = fmaxf(fmaf(pn, w1s[kcol],     b1s[kcol]),     0.0f) * qn;
      a.y = fmaxf(fmaf(pn, w1s[kcol + 1], b1s[kcol + 1]), 0.0f) * qn;
      v2f b; b.x = breg[kk][0]; b.y = breg[kk][1];
      acc = __builtin_amdgcn_wmma_f32_16x16x4_f32(
          false, a, false, b, (short)0, acc, false, false);
    }
#pragma unroll
    for (int k = 0; k < 8; ++k) {
      int row = i * 16 + k + 8 * lhi;
      tbuf[row * PITCH + wave * 16 + lmod] = acc[k];
    }
  }
  __syncthreads();

  // ---- phase 5: GEMM2: agg = Adj @ t  (densified edge scatter) -----------
  // K = 200 -> 50 k-steps; fused inv_in + bias + relu + mean-pool in regs.
  const float b2c = b2[wave * 16 + lmod];
  float psum = 0.0f;
  for (int i = 0; i < ROWSP / 16; ++i) {
    const float* arow = &adj[(i * 16 + lmod) * APITCH + 2 * lhi];
    const float* bcol = &tbuf[(2 * lhi) * PITCH + wave * 16 + lmod];
    v8f acc = {0.f, 0.f, 0.f, 0.f, 0.f, 0.f, 0.f, 0.f};
#pragma unroll
    for (int kk = 0; kk < 50; ++kk) {
      v2f a; a.x = arow[kk * 4];           a.y = arow[kk * 4 + 1];
      v2f b; b.x = bcol[(kk * 4) * PITCH]; b.y = bcol[(kk * 4 + 1) * PITCH];
      acc = __builtin_amdgcn_wmma_f32_16x16x4_f32(
          false, a, false, b, (short)0, acc, false, false);
    }
#pragma unroll
    for (int k = 0; k < 8; ++k) {
      int row = i * 16 + k + 8 * lhi;
      if (row < NODES) {
        float v = fmaf(acc[k], invIn[row], b2c);
        psum += fmaxf(v, 0.0f);
      }
    }
  }
  // combine the two lane halves (same column, disjoint row sets)
  psum += __shfl_xor(psum, 16);
  if (lane < 16)
    hg[g * HID + wave * 16 + lane] = psum * (1.0f / (float)NODES);
}

// ---------------------------------------------------------------------------
// Generic WMMA f32 GEMM: C[M x N] = act(A[M x K] @ W[K x N] + bias).
// One wave per 16x16 output tile, 8 waves per block. M%16==0, K%4==0.
// ---------------------------------------------------------------------------
__global__ __launch_bounds__(256) void wmma_gemm_bias_kernel(
    const float* __restrict__ A, const float* __restrict__ W,
    const float* __restrict__ bias, float* __restrict__ C,
    int M, int K, int Nn, int relu)
{
  const int lane = threadIdx.x & 31;
  const int wave = threadIdx.x >> 5;
  const int nTilesN = (Nn + 15) >> 4;
  const int nTilesM = M >> 4;
  const int tile = blockIdx.x * 8 + wave;
  if (tile >= nTilesM * nTilesN) return;
  const int tm = tile / nTilesN, tn = tile % nTilesN;
  const int lmod = lane & 15, lhi = lane >> 4;
  const int col  = tn * 16 + lmod;
  const int colc = (col < Nn) ? col : (Nn - 1);   // clamp for N=10 tail tile

  const float* arow = A + (size_t)(tm * 16 + lmod) * K + 2 * lhi;
  const float* bcol = W + (size_t)(2 * lhi) * Nn + colc;

  v8f acc = {0.f, 0.f, 0.f, 0.f, 0.f, 0.f, 0.f, 0.f};
  const int kq = K >> 2;
#pragma unroll 4
  for (int kk = 0; kk < kq; ++kk) {
    v2f a; a.x = arow[kk * 4];                a.y = arow[kk * 4 + 1];
    v2f b; b.x = bcol[(size_t)(kk * 4) * Nn]; b.y = bcol[(size_t)(kk * 4 + 1) * Nn];
    acc = __builtin_amdgcn_wmma_f32_16x16x4_f32(
        false, a, false, b, (short)0, acc, false, false);
  }

  const float bv = bias[colc];
#pragma unroll
  for (int k = 0; k < 8; ++k) {
    int row = tm * 16 + k + 8 * lhi;
    float v = acc[k] + bv;
    if (relu) v = fmaxf(v, 0.0f);
    if (col < Nn) C[(size_t)row * Nn + col] = v;
  }
}

// ---------------------------------------------------------------------------
__global__ __launch_bounds__(256) void softmax_rows_kernel(
    const float* __restrict__ in, float* __restrict__ out)
{
  int r = blockIdx.x * blockDim.x + threadIdx.x;
  if (r >= NGRAPH) return;
  float v[NCLS];
  float m = -3.4e38f;
  for (int c = 0; c < NCLS; ++c) { v[c] = in[r * NCLS + c]; m = fmaxf(m, v[c]); }
  float s = 0.0f;
  for (int c = 0; c < NCLS; ++c) { v[c] = __expf(v[c] - m); s += v[c]; }
  float inv = 1.0f / s;
  for (int c = 0; c < NCLS; ++c) out[r * NCLS + c] = v[c] * inv;
}

// ---------------------------------------------------------------------------
extern "C" void kernel_launch(void* const* d_in, const int* in_sizes, int n_in,
                              void* d_out, int out_size, void* d_ws, size_t ws_size,
                              hipStream_t stream) {
  const int*   src = (const int*)d_in[0];
  const int*   dst = (const int*)d_in[1];
  const float* W1  = (const float*)d_in[2];
  const float* b1  = (const float*)d_in[3];
  const float* W2  = (const float*)d_in[4];
  const float* b2  = (const float*)d_in[5];
  const float* Wa  = (const float*)d_in[6];
  const float* ba  = (const float*)d_in[7];
  const float* Wb  = (const float*)d_in[8];
  const float* bb  = (const float*)d_in[9];
  const float* Wc  = (const float*)d_in[10];
  const float* bc  = (const float*)d_in[11];
  const float* Wd  = (const float*)d_in[12];
  const float* bd  = (const float*)d_in[13];
  const float* We  = (const float*)d_in[14];
  const float* be  = (const float*)d_in[15];

  float* ws = (float*)d_ws;
  float* hg     = ws;                       // 512*128
  float* x1     = hg  + NGRAPH * HID;       // 512*512
  float* x2     = x1  + NGRAPH * 512;       // 512*1024
  float* x3     = x2  + NGRAPH * 1024;      // 512*1024
  float* x4     = x3  + NGRAPH * 1024;      // 512*512
  float* logits = x4  + NGRAPH * 512;       // 512*10

  // GNN stage: one workgroup per graph, everything in LDS, scatter densified
  gnn_fused_kernel<<<NGRAPH, 256, 0, stream>>>(src, dst, W1, b1, W2, b2, hg);

  // MLP head: blocks = ceil(tilesM*tilesN / 8), tilesM = 512/16 = 32
  wmma_gemm_bias_kernel<<<(32 * 32 + 7) / 8, 256, 0, stream>>>(hg, Wa, ba, x1, NGRAPH, 128, 512, 1);
  wmma_gemm_bias_kernel<<<(32 * 64 + 7) / 8, 256, 0, stream>>>(x1, Wb, bb, x2, NGRAPH, 512, 1024, 1);
  wmma_gemm_bias_kernel<<<(32 * 64 + 7) / 8, 256, 0, stream>>>(x2, Wc, bc, x3, NGRAPH, 1024, 1024, 1);
  wmma_gemm_bias_kernel<<<(32 * 32 + 7) / 8, 256, 0, stream>>>(x3, Wd, bd, x4, NGRAPH, 1024, 512, 1);
  wmma_gemm_bias_kernel<<<(32 * 1 + 7) / 8, 256, 0, stream>>>(x4, We, be, logits, NGRAPH, 512, 10, 0);

  softmax_rows_kernel<<<(NGRAPH + 255) / 256, 256, 0, stream>>>(logits, (float*)d_out);
}